// LogLoss_42958262894786
// MI455X (gfx1250) — compile-verified
//
#include <hip/hip_runtime.h>

// Problem constants (match reference)
constexpr int   kN    = 524288;
constexpr int   kC    = 128;
constexpr float kClip = 1e-10f;

constexpr int kBlock         = 256;                     // 8 wave32 per block
constexpr int kRowsPerThread = 8;                       // 8 gathers in flight / thread
constexpr int kBlocks        = kN / (kBlock * kRowsPerThread); // 256
constexpr int kTotalThreads  = kBlock * kBlocks;        // 65536

// ---------------------------------------------------------------------------
// Kernel 1: per-block partial sums of weight[t] * -log(max(x, clip))
// ---------------------------------------------------------------------------
__global__ __launch_bounds__(kBlock) void logloss_partial_kernel(
    const float* __restrict__ input,    // (N, C) row-major
    const int*   __restrict__ target,   // (N,)
    const float* __restrict__ weight,   // (C,)
    float*       __restrict__ partials) // (kBlocks,)
{
    __shared__ float s_w[kC];
    __shared__ float s_red[kBlock / 32];

    const int tid = threadIdx.x;

    // Stage the tiny weight table in LDS (one ds_load per use later).
    if (tid < kC) s_w[tid] = weight[tid];
    __syncthreads();

    const int gtid = blockIdx.x * kBlock + tid;

    // Speculatively prefetch each row's base line while the (dependent)
    // target loads are still in flight. Locality 3 -> WGP scope: pull
    // through all cache levels so the later gather can hit near.
#pragma unroll
    for (int k = 0; k < kRowsPerThread; ++k) {
        const size_t row = (size_t)gtid + (size_t)k * kTotalThreads;
        __builtin_prefetch(input + row * kC, 0, 3);
    }

    // Coalesced target loads: for each k, lane L reads target[base + L].
    int t[kRowsPerThread];
#pragma unroll
    for (int k = 0; k < kRowsPerThread; ++k) {
        t[k] = target[gtid + k * kTotalThreads];
    }

    // Independent gathers — 8 loads in flight per thread before any use.
    float v[kRowsPerThread];
#pragma unroll
    for (int k = 0; k < kRowsPerThread; ++k) {
        const size_t row = (size_t)gtid + (size_t)k * kTotalThreads;
        v[k] = input[row * kC + (size_t)t[k]];
    }

    float acc = 0.0f;
#pragma unroll
    for (int k = 0; k < kRowsPerThread; ++k) {
        const float x = fmaxf(v[k], kClip);   // clip(input, 1e-10)
        acc += s_w[t[k]] * (-logf(x));        // precise log, matches jnp.log
    }

    // Wave32 shuffle reduction.
#pragma unroll
    for (int off = 16; off > 0; off >>= 1) {
        acc += __shfl_down(acc, off);
    }
    if ((tid & 31) == 0) s_red[tid >> 5] = acc;
    __syncthreads();

    if (tid == 0) {
        float a = 0.0f;
#pragma unroll
        for (int w = 0; w < kBlock / 32; ++w) a += s_red[w];
        partials[blockIdx.x] = a;
    }
}

// ---------------------------------------------------------------------------
// Kernel 2: reduce block partials in double, write mean to d_out[0]
// ---------------------------------------------------------------------------
__global__ __launch_bounds__(256) void logloss_final_kernel(
    const float* __restrict__ partials,
    float*       __restrict__ out)
{
    __shared__ double s[256];
    double a = 0.0;
    for (int i = threadIdx.x; i < kBlocks; i += 256) {
        a += (double)partials[i];
    }
    s[threadIdx.x] = a;
    __syncthreads();

#pragma unroll
    for (int off = 128; off > 0; off >>= 1) {
        if (threadIdx.x < off) s[threadIdx.x] += s[threadIdx.x + off];
        __syncthreads();
    }
    if (threadIdx.x == 0) {
        out[0] = (float)(s[0] / (double)kN);
    }
}

// ---------------------------------------------------------------------------
extern "C" void kernel_launch(void* const* d_in, const int* in_sizes, int n_in,
                              void* d_out, int out_size, void* d_ws, size_t ws_size,
                              hipStream_t stream) {
    const float* input  = (const float*)d_in[0];  // (N, C) float32
    const int*   target = (const int*)  d_in[1];  // (N,)   int
    const float* weight = (const float*)d_in[2];  // (C,)   float32
    float*       out    = (float*)d_out;

    float* partials = (float*)d_ws;               // kBlocks floats of scratch

    logloss_partial_kernel<<<kBlocks, kBlock, 0, stream>>>(input, target, weight, partials);
    logloss_final_kernel<<<1, 256, 0, stream>>>(partials, out);
}